// GNN_4844723110524
// MI455X (gfx1250) — compile-verified
//
#include <hip/hip_runtime.h>
#include <hip/hip_bf16.h>

// ---------------- WMMA types ----------------
typedef __attribute__((ext_vector_type(2))) float v2f;
typedef __attribute__((ext_vector_type(8))) float v8f;

#define N_NODES 100000
#define N_EDGES 3200000

// ---------------- degree / norm ----------------
__global__ void deg_init_kernel(float* deg, int n) {
    int i = blockIdx.x * blockDim.x + threadIdx.x;
    if (i < n) deg[i] = 1.0f;  // self-loop
}
__global__ void deg_count_kernel(const int* __restrict__ dst, float* deg, int e) {
    int i = blockIdx.x * blockDim.x + threadIdx.x;
    if (i < e) atomicAdd(&deg[dst[i]], 1.0f);
}
__global__ void deg_rsqrt_kernel(float* deg, int n) {
    int i = blockIdx.x * blockDim.x + threadIdx.x;
    if (i < n) deg[i] = rsqrtf(deg[i]);  // deg >= 1 always
}

// ---------------- fp32 GEMM via V_WMMA_F32_16X16X4_F32 ----------------
// C[M x NOUT] = A[M x K] * B[K x NOUT].  One 16x16 output tile per wave.
// Block: MT m-tiles x (NOUT/16) n-tiles = MT*NT waves; 16*MT rows per block.
template <int K, int NOUT, int MT>
__global__ __launch_bounds__(MT*(NOUT/16)*32)
void gemm_wmma_f32(const float* __restrict__ A, const float* __restrict__ B,
                   float* __restrict__ C, int M) {
    constexpr int NT   = NOUT / 16;
    constexpr int NW   = MT * NT;
    constexpr int ROWS = MT * 16;
    constexpr int KP   = K + 4;     // pad -> conflict-free A reads
    constexpr int NP   = NOUT + 4;  // pad -> conflict-free B reads

    __shared__ float As[ROWS][KP];
    __shared__ float Bs[K][NP];

    const int tid  = threadIdx.x;
    const int row0 = blockIdx.x * ROWS;

    // Cooperative float4 staging (uniform loops; no branch divergence).
    constexpr int A_ELTS = ROWS * K;
    for (int i = tid * 4; i < A_ELTS; i += NW * 32 * 4) {
        int r = i / K, c = i % K;
        int gr = row0 + r;
        float4 v = (gr < M) ? *(const float4*)(A + (long long)gr * K + c)
                            : make_float4(0.f, 0.f, 0.f, 0.f);
        *(float4*)(&As[r][c]) = v;
    }
    constexpr int B_ELTS = K * NOUT;
    for (int i = tid * 4; i < B_ELTS; i += NW * 32 * 4) {
        int r = i / NOUT, c = i % NOUT;
        *(float4*)(&Bs[r][c]) = *(const float4*)(B + i);
    }
    __syncthreads();

    const int wid  = tid >> 5;
    const int lane = tid & 31;
    const int mn   = lane & 15;   // M index for A-frag, N index for B/C/D-frag
    const int h    = lane >> 4;   // half-wave select
    const int mt   = wid / NT;
    const int nt   = wid % NT;

    v8f acc = {};
    const float* arow = &As[mt * 16 + mn][0];
    const int    ncol = nt * 16 + mn;

#pragma unroll
    for (int k = 0; k < K; k += 4) {
        // A 16x4 f32 frag: VGPR j holds K = k + j + 2*h   (ISA 7.12.2)
        v2f a, b;
        a.x = arow[k + 2 * h + 0];
        a.y = arow[k + 2 * h + 1];
        // B 4x16 frag (row-striped across lanes): VGPR j holds K-row k + j + 2*h
        b.x = Bs[k + 2 * h + 0][ncol];
        b.y = Bs[k + 2 * h + 1][ncol];
        acc = __builtin_amdgcn_wmma_f32_16x16x4_f32(
            /*neg_a=*/false, a, /*neg_b=*/false, b,
            /*c_mod=*/(short)0, acc, /*reuse_a=*/false, /*reuse_b=*/false);
    }

    // D 16x16 f32: VGPR j -> row (j + 8*h), lane -> col
#pragma unroll
    for (int j = 0; j < 8; ++j) {
        int r = row0 + mt * 16 + j + 8 * h;
        if (r < M) C[(long long)r * NOUT + ncol] = acc[j];
    }
}

// ---------------- zero ----------------
__global__ void zero_kernel(float* p, long long n) {
    long long i = (long long)blockIdx.x * blockDim.x + threadIdx.x;
    if (i < n) p[i] = 0.0f;
}

// ---------------- edge scatter:  agg[dst] += h[src] * dinv[src]*dinv[dst] ----------------
template <int F>
__global__ void scatter_kernel(const float* __restrict__ h, const int* __restrict__ src,
                               const int* __restrict__ dst, const float* __restrict__ dinv,
                               float* agg, int E) {
    constexpr int LPE = F / 4;  // lanes per edge, float4 each
    long long t = (long long)blockIdx.x * blockDim.x + threadIdx.x;
    int e = (int)(t / LPE);
    int j = (int)(t % LPE);
    if (e >= E) return;
    int s = src[e], d = dst[e];
    float norm = dinv[s] * dinv[d];
    float4 v = *(const float4*)(h + (long long)s * F + j * 4);
    float* o = agg + (long long)d * F + j * 4;
    atomicAdd(o + 0, v.x * norm);
    atomicAdd(o + 1, v.y * norm);
    atomicAdd(o + 2, v.z * norm);
    atomicAdd(o + 3, v.w * norm);
}

// ---------------- finish: y = relu(agg + h*dinv^2 + b)  (in place over agg) ----------------
template <int F>
__global__ void finish_kernel(float* agg, const float* __restrict__ h,
                              const float* __restrict__ dinv, const float* __restrict__ bias,
                              int n) {
    long long t = (long long)blockIdx.x * blockDim.x + threadIdx.x;
    int i = (int)(t / F);
    int f = (int)(t % F);
    if (i >= n) return;
    float di = dinv[i];
    float v = agg[(long long)i * F + f] + h[(long long)i * F + f] * di * di + bias[f];
    agg[(long long)i * F + f] = fmaxf(v, 0.0f);
}

// ---------------- final linear: out = y2 @ Wl + bl ----------------
__global__ void final_linear_kernel(const float* __restrict__ y, const float* __restrict__ Wl,
                                    const float* __restrict__ bl, float* out, int n) {
    int i = blockIdx.x * blockDim.x + threadIdx.x;
    if (i >= n) return;
    const float* row = y + (long long)i * 32;
    float s = 0.0f;
#pragma unroll
    for (int f = 0; f < 32; ++f) s += row[f] * Wl[f];
    out[i] = s + bl[0];
}

static inline size_t align256(size_t x) { return (x + 255) & ~size_t(255); }

extern "C" void kernel_launch(void* const* d_in, const int* in_sizes, int n_in,
                              void* d_out, int out_size, void* d_ws, size_t ws_size,
                              hipStream_t stream) {
    const float* x    = (const float*)d_in[0];           // [N,128]
    const int*   ei   = (const int*)d_in[1];             // [2,E]
    const float* W1   = (const float*)d_in[2];           // [128,64]
    const float* b1   = (const float*)d_in[3];           // [64]
    const float* W2   = (const float*)d_in[4];           // [64,32]
    const float* b2   = (const float*)d_in[5];           // [32]
    const float* Wl   = (const float*)d_in[6];           // [32,1]
    const float* bl   = (const float*)d_in[7];           // [1]
    float*       out  = (float*)d_out;                   // [N,1]

    const int N = N_NODES;
    const int E = N_EDGES;
    const int* src = ei;
    const int* dst = ei + E;

    // Workspace layout
    char* ws = (char*)d_ws;
    size_t off = 0;
    float* dinv = (float*)(ws + off); off += align256((size_t)N * 4);
    float* h1   = (float*)(ws + off); off += align256((size_t)N * 64 * 4);
    float* agg1 = (float*)(ws + off); off += align256((size_t)N * 64 * 4);  // becomes y1
    float* h2   = (float*)(ws + off); off += align256((size_t)N * 32 * 4);
    float* agg2 = (float*)(ws + off); off += align256((size_t)N * 32 * 4);  // becomes y2
    (void)ws_size; (void)n_in; (void)in_sizes; (void)out_size;

    const int T = 256;

    // 1) symmetric-normalization coefficients
    deg_init_kernel<<<(N + T - 1) / T, T, 0, stream>>>(dinv, N);
    deg_count_kernel<<<(E + T - 1) / T, T, 0, stream>>>(dst, dinv, E);
    deg_rsqrt_kernel<<<(N + T - 1) / T, T, 0, stream>>>(dinv, N);

    // 2) h1 = x @ W1   (K=128, NOUT=64, 32 rows/block -> 3125 blocks, 256 thr)
    gemm_wmma_f32<128, 64, 2><<<N / 32, 2 * 4 * 32, 0, stream>>>(x, W1, h1, N);

    // 3) layer-1 aggregation
    zero_kernel<<<(int)(((long long)N * 64 + T - 1) / T), T, 0, stream>>>(agg1, (long long)N * 64);
    {
        long long tt = (long long)E * (64 / 4);
        scatter_kernel<64><<<(int)((tt + T - 1) / T), T, 0, stream>>>(h1, src, dst, dinv, agg1, E);
    }
    finish_kernel<64><<<(int)(((long long)N * 64 + T - 1) / T), T, 0, stream>>>(agg1, h1, dinv, b1, N);

    // 4) h2 = y1 @ W2  (K=64, NOUT=32, 32 rows/block -> 3125 blocks, 128 thr)
    gemm_wmma_f32<64, 32, 2><<<N / 32, 2 * 2 * 32, 0, stream>>>(agg1, W2, h2, N);

    // 5) layer-2 aggregation
    zero_kernel<<<(int)(((long long)N * 32 + T - 1) / T), T, 0, stream>>>(agg2, (long long)N * 32);
    {
        long long tt = (long long)E * (32 / 4);
        scatter_kernel<32><<<(int)((tt + T - 1) / T), T, 0, stream>>>(h2, src, dst, dinv, agg2, E);
    }
    finish_kernel<32><<<(int)(((long long)N * 32 + T - 1) / T), T, 0, stream>>>(agg2, h2, dinv, b2, N);

    // 6) out = y2 @ Wl + bl
    final_linear_kernel<<<(N + T - 1) / T, T, 0, stream>>>(agg2, Wl, bl, out, N);
}